// ElectronicDensityLayer_7988639170995
// MI455X (gfx1250) — compile-verified
//
#include <hip/hip_runtime.h>
#include <math.h>

// ---------------------------------------------------------------------------
// ElectronicDensityLayer for MI455X (gfx1250)
//  - scatter: memory/atomic bound, plain VALU + global_atomic_add_f32
//  - spectral solve: 4x FP32 GEMM (512^3) via V_WMMA_F32_16X16X4_F32
// ---------------------------------------------------------------------------

typedef __attribute__((ext_vector_type(2))) float v2f;
typedef __attribute__((ext_vector_type(8))) float v8f;

#define MX 512
#define NYB 512
#define NELEM (MX * NYB)
#define PI_F 3.14159265358979323846f
#define SQRT2_F 1.41421356237309504880f

// ---------------------------------------------------------------------------
// Init: DM = init_density_map; build C (DCT cos table), IC = w[k]*C,
// PSCALE = 1/(wj^2+wk^2) * 1/(MX*NYB) with [0][0] = 0; zero out[0].
// ---------------------------------------------------------------------------
__global__ __launch_bounds__(256) void edl_init_kernel(
    const float* __restrict__ init_dm, const float* __restrict__ unit_len,
    float* __restrict__ DM, float* __restrict__ C, float* __restrict__ IC,
    float* __restrict__ PSCALE, float* __restrict__ out)
{
    int idx = blockIdx.x * blockDim.x + threadIdx.x;
    if (idx == 0) out[0] = 0.0f;
    if (idx >= NELEM) return;

    int k = idx >> 9;       // row   (0..511)
    int m = idx & 511;      // col   (0..511)

    DM[idx] = init_dm[idx];

    // DCT-II basis: C[k][m] = cos(pi * k * (2m+1) / (2n)), n = 512 (MX == NYB)
    float ang = PI_F * (float)k * (2.0f * (float)m + 1.0f) * (1.0f / (2.0f * (float)MX));
    float c = cosf(ang);
    C[idx] = c;
    IC[idx] = (k == 0 ? 1.0f : 2.0f) * c;

    // pscale with 1/(MX*NYB) folded in
    float ratio = unit_len[0] / unit_len[1];
    float wj = (float)k * (2.0f * PI_F / (float)MX);
    float wk = (float)m * (2.0f * PI_F / (float)NYB) * ratio;
    float wsq = wj * wj + wk * wk;
    float ps = (idx == 0) ? 0.0f : (1.0f / wsq);
    PSCALE[idx] = ps * (1.0f / ((float)MX * (float)NYB));
}

// ---------------------------------------------------------------------------
// pre_normalize + 4x4 bilinear-overlap scatter with global f32 atomics.
// ---------------------------------------------------------------------------
__global__ __launch_bounds__(256) void edl_scatter_kernel(
    const float* __restrict__ pos, const float* __restrict__ size,
    const float* __restrict__ wgt, const float* __restrict__ expr,
    const float* __restrict__ unit_len, float* __restrict__ DM, int n)
{
    int i = blockIdx.x * blockDim.x + threadIdx.x;
    if (i >= n) return;

    float ux = unit_len[0], uy = unit_len[1];
    float px = pos[2 * i + 0], py = pos[2 * i + 1];
    float w0 = size[2 * i + 0], h0 = size[2 * i + 1];

    float nw = fmaxf(w0, SQRT2_F * ux);
    float nh = fmaxf(h0, SQRT2_F * uy);
    float wt = wgt[i] * expr[i] * (w0 * h0) / (nw * nh);

    float inv_ux = 1.0f / ux, inv_uy = 1.0f / uy;
    float xl = fminf(fmaxf((px - 0.5f * nw) * inv_ux, 0.0f), (float)MX);
    float xh = fminf(fmaxf((px + 0.5f * nw) * inv_ux, 0.0f), (float)MX);
    float yl = fminf(fmaxf((py - 0.5f * nh) * inv_uy, 0.0f), (float)NYB);
    float yh = fminf(fmaxf((py + 0.5f * nh) * inv_uy, 0.0f), (float)NYB);

    int bx0 = (int)floorf(xl);
    int by0 = (int)floorf(yl);

    #pragma unroll
    for (int dx = 0; dx < 4; ++dx) {
        int ix = bx0 + dx;
        float fx = (float)ix;
        float ox = fmaxf(fminf(xh, fx + 1.0f) - fmaxf(xl, fx), 0.0f);
        int ixc = min(max(ix, 0), MX - 1);
        float wox = wt * ox;
        #pragma unroll
        for (int dy = 0; dy < 4; ++dy) {
            int iy = by0 + dy;
            float fy = (float)iy;
            float oy = fmaxf(fminf(yh, fy + 1.0f) - fmaxf(yl, fy), 0.0f);
            int iyc = min(max(iy, 0), NYB - 1);
            atomicAdd(&DM[ixc * NYB + iyc], wox * oy);
        }
    }
}

// ---------------------------------------------------------------------------
// 512x512x512 FP32 GEMM: D = op(A) * op(B), one 16x16 tile per wave32.
// Uses V_WMMA_F32_16X16X4_F32. A fragment (16x4): lanes 0-15 hold K=k0,k0+1;
// lanes 16-31 hold K=k0+2,k0+3 (VGPR0/1). B fragment (4x16) mirrors with N on
// lanes. C/D: v8f, lanes 0-15 -> rows tm..tm+7, lanes 16-31 -> rows tm+8..+15.
// ---------------------------------------------------------------------------
template <bool TA, bool TB>
__global__ __launch_bounds__(256) void wmma_gemm512(
    const float* __restrict__ A, const float* __restrict__ B,
    float* __restrict__ D)
{
    const int lane = threadIdx.x & 31;
    const int wave = blockIdx.x * (blockDim.x >> 5) + (threadIdx.x >> 5);
    const int tm = (wave >> 5) << 4;   // tile row origin
    const int tn = (wave & 31) << 4;   // tile col origin
    const int half = lane >> 4;        // 0: K pair {0,1}; 1: K pair {2,3}
    const int l15 = lane & 15;

    v8f acc = {};

    for (int k0 = 0; k0 < 512; k0 += 4) {
        const int kb = k0 + 2 * half;
        v2f a, b;

        if (TA) {
            // op(A)[m][k] = A[k][m]  (strided loads)
            a.x = A[(kb + 0) * 512 + tm + l15];
            a.y = A[(kb + 1) * 512 + tm + l15];
        } else {
            // row-major contiguous pair, 8B aligned (kb even)
            a = *(const v2f*)&A[(tm + l15) * 512 + kb];
        }

        if (TB) {
            // op(B)[k][n] = B[n][k]  (contiguous pair per lane)
            b = *(const v2f*)&B[(tn + l15) * 512 + kb];
        } else {
            b.x = B[(kb + 0) * 512 + tn + l15];
            b.y = B[(kb + 1) * 512 + tn + l15];
        }

        acc = __builtin_amdgcn_wmma_f32_16x16x4_f32(
            /*neg_a=*/false, a, /*neg_b=*/false, b,
            /*c_mod=*/(short)0, acc, /*reuse_a=*/false, /*reuse_b=*/false);
    }

    #pragma unroll
    for (int v = 0; v < 8; ++v) {
        D[(tm + v + 8 * half) * 512 + tn + l15] = acc[v];
    }
}

// ---------------------------------------------------------------------------
// coeff *= pscale (elementwise, in place)
// ---------------------------------------------------------------------------
__global__ __launch_bounds__(256) void edl_scale_kernel(
    float* __restrict__ COEFF, const float* __restrict__ PSCALE)
{
    int idx = blockIdx.x * blockDim.x + threadIdx.x;
    if (idx < NELEM) COEFF[idx] *= PSCALE[idx];
}

// ---------------------------------------------------------------------------
// energy = sum(POT * DM); block reduce + atomicAdd into out[0]
// ---------------------------------------------------------------------------
__global__ __launch_bounds__(256) void edl_energy_kernel(
    const float* __restrict__ POT, const float* __restrict__ DM,
    float* __restrict__ out)
{
    __shared__ float sdata[256];
    int tid = threadIdx.x;
    float s = 0.0f;
    for (int i = blockIdx.x * 256 + tid; i < NELEM; i += gridDim.x * 256)
        s += POT[i] * DM[i];
    sdata[tid] = s;
    __syncthreads();
    for (int off = 128; off > 0; off >>= 1) {
        if (tid < off) sdata[tid] += sdata[tid + off];
        __syncthreads();
    }
    if (tid == 0) atomicAdd(out, sdata[0]);
}

// ---------------------------------------------------------------------------
// Launch sequence (single stream, graph-capture safe)
// ---------------------------------------------------------------------------
extern "C" void kernel_launch(void* const* d_in, const int* in_sizes, int n_in,
                              void* d_out, int out_size, void* d_ws, size_t ws_size,
                              hipStream_t stream)
{
    const float* node_pos    = (const float*)d_in[0];
    const float* node_size   = (const float*)d_in[1];
    const float* node_weight = (const float*)d_in[2];
    const float* expand      = (const float*)d_in[3];
    const float* unit_len    = (const float*)d_in[4];
    const float* init_dm     = (const float*)d_in[5];
    // d_in[6], d_in[7]: num_bin_x/y == 512, compile-time constants here.

    const int n_nodes = in_sizes[0] / 2;
    float* out = (float*)d_out;

    // Workspace layout: 6 x 512x512 f32 buffers (6 MB total)
    float* ws     = (float*)d_ws;
    float* DM     = ws + 0 * NELEM;
    float* C      = ws + 1 * NELEM;
    float* IC     = ws + 2 * NELEM;
    float* PSCALE = ws + 3 * NELEM;
    float* BUF1   = ws + 4 * NELEM;   // T1, then P1
    float* BUF2   = ws + 5 * NELEM;   // coeff, then potential

    // 1) tables + DM init + out zero
    edl_init_kernel<<<(NELEM + 255) / 256, 256, 0, stream>>>(
        init_dm, unit_len, DM, C, IC, PSCALE, out);

    // 2) density scatter
    edl_scatter_kernel<<<(n_nodes + 255) / 256, 256, 0, stream>>>(
        node_pos, node_size, node_weight, expand, unit_len, DM, n_nodes);

    // 3) spectral solve as 4 FP32 WMMA GEMMs
    //    T1    = C  @ DM        (NN)
    //    coeff = T1 @ C^T       (NT)
    //    coeff *= pscale/(MN)
    //    P1    = coeff @ IC     (NN)
    //    POT   = IC^T @ P1      (TN)
    wmma_gemm512<false, false><<<128, 256, 0, stream>>>(C,    DM,   BUF1);
    wmma_gemm512<false, true ><<<128, 256, 0, stream>>>(BUF1, C,    BUF2);
    edl_scale_kernel<<<(NELEM + 255) / 256, 256, 0, stream>>>(BUF2, PSCALE);
    wmma_gemm512<false, false><<<128, 256, 0, stream>>>(BUF2, IC,   BUF1);
    wmma_gemm512<true,  false><<<128, 256, 0, stream>>>(IC,   BUF1, BUF2);

    // 4) energy = sum(potential * density)
    edl_energy_kernel<<<256, 256, 0, stream>>>(BUF2, DM, out);
}